// ASD_78975858639065
// MI455X (gfx1250) — compile-verified
//
#include <hip/hip_runtime.h>

// ---------------- problem constants ----------------
#define NN     50000
#define MPAD   50048            // 782 * 64, row-padded node count
#define NB     128
#define ND     35
#define NT     140
#define E_IN   300000
#define E_EX   100000
#define EPSBN  1e-5f
#define OBLD   9808             // fc1 input row stride (9800 padded to 16)

typedef float v2f __attribute__((ext_vector_type(2)));
typedef float v8f __attribute__((ext_vector_type(8)));

#if __has_builtin(__builtin_amdgcn_global_load_async_to_lds_b128) && \
    __has_builtin(__builtin_amdgcn_s_wait_asynccnt)
#define USE_ASYNC_LDS 1
typedef int vi4 __attribute__((ext_vector_type(4)));
typedef __attribute__((address_space(1))) vi4 as1_vi4;   // global int4*
typedef __attribute__((address_space(3))) vi4 as3_vi4;   // LDS int4*
#endif

// ---------------- fp32 WMMA GEMM, LDS-tiled (A and B staged) ----------------
// C[M,N] = A[M,Kreal] @ B[Kreal,N] (+bias) (optional relu), row-major.
// A has leading dim Kpad (%16==0) with ZERO padding in cols [Kreal,Kpad).
// M % 64 == 0. RAGGED=false requires N % 64 == 0 (guard-free stores).
// RAGGED=true: N arbitrary; B loads index-clamped (garbage reaches only dead
// lanes whose stores are guarded). No exec-mask branches in the WMMA stream.
template <bool RAGGED>
__global__ __launch_bounds__(128) void gemm_wmma_tiled(
    const float* __restrict__ A, const float* __restrict__ Bw,
    const float* __restrict__ bias, float* __restrict__ C,
    int M, int Kpad, int Kreal, int N, int relu)
{
  __shared__ __align__(16) float sB[16 * 64];   // sB[k][col]
  __shared__ __align__(16) float sA[16 * 64];   // sA[k][row] (transposed tile)

  const int wave = threadIdx.x >> 5;
  const int lane = threadIdx.x & 31;
  const int hlf  = lane >> 4;      // K-pair select per ISA A/B layout
  const int l    = lane & 15;
  const int rowL = wave * 16 + l;  // row within 64-row block tile
  const int colBase = blockIdx.x * 64;

  // cooperative B-load decomposition: 128 thr x 8 elem = 16x64 tile
  const int ldr  = threadIdx.x >> 3;    // tile row 0..15
  const int lseg = threadIdx.x & 7;     // 8-wide column segment
  // cooperative A-load decomposition: 64 rows x 16 k
  const int arow = threadIdx.x >> 1;          // 0..63
  const int akh  = (threadIdx.x & 1) * 8;     // k-half 0 or 8
  const float* aSrc = A + (size_t)(blockIdx.y * 64 + arow) * Kpad + akh;

  v8f acc[4] = {};

  for (int kc = 0; kc < Kpad; kc += 16) {
    // ---- stage B tile rows [kc,kc+16) x cols [colBase,colBase+64) ----
    {
      const int gk = kc + ldr;
      float* dstp = &sB[ldr * 64 + lseg * 8];
      if (!RAGGED) {
        if (gk < Kreal) {
          const float* gp = Bw + (size_t)gk * N + colBase + lseg * 8;
#ifdef USE_ASYNC_LDS
          __builtin_amdgcn_global_load_async_to_lds_b128(
              (as1_vi4*)gp, (as3_vi4*)dstp, 0, 0);
          __builtin_amdgcn_global_load_async_to_lds_b128(
              (as1_vi4*)(gp + 4), (as3_vi4*)(dstp + 4), 0, 0);
#else
          ((float4*)dstp)[0] = ((const float4*)gp)[0];
          ((float4*)dstp)[1] = ((const float4*)gp)[1];
#endif
        } else {
#pragma unroll
          for (int i = 0; i < 8; ++i) dstp[i] = 0.f;
        }
      } else {
        const size_t lastB = (size_t)Kreal * N - 1;
#pragma unroll
        for (int i = 0; i < 8; ++i) {
          size_t fidx = (size_t)gk * N + colBase + lseg * 8 + i;
          if (fidx > lastB) fidx = lastB;       // v_min, not a branch
          float val = Bw[fidx];
          if (gk >= Kreal) val = 0.f;           // zero K tail
          dstp[i] = val;
        }
      }
      __builtin_prefetch(Bw + (size_t)(gk + 16) * N + colBase + lseg * 8, 0, 3);
    }
    // ---- stage A tile (transposed) rows x k: sA[k][row] ----
    {
      const float4 a0 = *(const float4*)(aSrc + kc);
      const float4 a1 = *(const float4*)(aSrc + kc + 4);
      sA[(akh + 0) * 64 + arow] = a0.x;
      sA[(akh + 1) * 64 + arow] = a0.y;
      sA[(akh + 2) * 64 + arow] = a0.z;
      sA[(akh + 3) * 64 + arow] = a0.w;
      sA[(akh + 4) * 64 + arow] = a1.x;
      sA[(akh + 5) * 64 + arow] = a1.y;
      sA[(akh + 6) * 64 + arow] = a1.z;
      sA[(akh + 7) * 64 + arow] = a1.w;
      __builtin_prefetch(aSrc + kc + 16, 0, 3);
    }
#ifdef USE_ASYNC_LDS
    if (!RAGGED) __builtin_amdgcn_s_wait_asynccnt(0);
#endif
    __syncthreads();

    // ---- 4 WMMA K-steps per chunk, LDS-only operands ----
#pragma unroll
    for (int kk = 0; kk < 16; kk += 4) {
      const int k0 = kk + 2 * hlf;
      v2f a, b;
      a.x = sA[(k0 + 0) * 64 + rowL];
      a.y = sA[(k0 + 1) * 64 + rowL];
#pragma unroll
      for (int j = 0; j < 4; ++j) {
        const int tcol = j * 16 + l;
        b.x = sB[(k0 + 0) * 64 + tcol];
        b.y = sB[(k0 + 1) * 64 + tcol];
        acc[j] = __builtin_amdgcn_wmma_f32_16x16x4_f32(
            false, a, false, b, (short)0, acc[j], false, false);
      }
    }
    __syncthreads();
  }

  // C/D layout: lanes 0-15 -> N=l, VGPR r -> M=r; lanes 16-31 -> M=r+8
  const int rowBase = blockIdx.y * 64 + wave * 16;
#pragma unroll
  for (int j = 0; j < 4; ++j) {
    const int col = colBase + j * 16 + l;
    if (!RAGGED || col < N) {
      const float bv = bias ? bias[col] : 0.f;
#pragma unroll
      for (int r = 0; r < 8; ++r) {
        const int row = rowBase + r + 8 * hlf;
        float val = acc[j][r] + bv;
        if (relu) val = fmaxf(val, 0.f);
        C[(size_t)row * N + col] = val;
      }
    }
  }
}

// ---------------- small utility kernels ----------------
__global__ void zero_kernel(float* __restrict__ p, long long n) {
  long long i = blockIdx.x * (long long)blockDim.x + threadIdx.x;
  if (i < n) p[i] = 0.f;
}

// dst[r*ldDst + c] = src[r*cols + c]
__global__ void copy_pad_kernel(float* __restrict__ dst, const float* __restrict__ src,
                                long long rows, int cols, int ldDst) {
  long long idx = blockIdx.x * (long long)blockDim.x + threadIdx.x;
  if (idx >= rows * cols) return;
  long long r = idx / cols;
  int c = (int)(idx % cols);
  dst[r * ldDst + c] = src[idx];
}

__global__ void init_deg_kernel(float* __restrict__ deg, int n) {
  int i = blockIdx.x * blockDim.x + threadIdx.x;
  if (i < n) deg[i] = 1.f;   // self loop
}

__global__ void count_deg_kernel(float* __restrict__ deg,
                                 const int* __restrict__ dst, int e) {
  int i = blockIdx.x * blockDim.x + threadIdx.x;
  if (i < e) atomicAdd(&deg[dst[i]], 1.f);
}

__global__ void finish_dinv_kernel(float* __restrict__ deg, int n) {
  int i = blockIdx.x * blockDim.x + threadIdx.x;
  if (i < n) deg[i] = rsqrtf(deg[i]);
}

// ---------------- GCN edge scatter (ld-aware) ----------------
__global__ void scatter_kernel(const float* __restrict__ hlin,
                               float* __restrict__ agg,
                               const int* __restrict__ src,
                               const int* __restrict__ dst,
                               const float* __restrict__ dinv,
                               long long total, int cols, int ld) {
  long long idx = blockIdx.x * (long long)blockDim.x + threadIdx.x;
  if (idx >= total) return;
  int e = (int)(idx / cols), f = (int)(idx % cols);
  int s = src[e], d = dst[e];
  float w = dinv[s] * dinv[d];
  atomicAdd(&agg[(size_t)d * ld + f], hlin[(size_t)s * ld + f] * w);
}

// agg = agg + hlin*dinv^2 + bias (optional relu), in place
__global__ void combine_kernel(float* __restrict__ agg,
                               const float* __restrict__ hlin,
                               const float* __restrict__ dinv,
                               const float* __restrict__ bias,
                               long long n, int cols, int ld, int relu) {
  long long idx = blockIdx.x * (long long)blockDim.x + threadIdx.x;
  if (idx >= n) return;
  int i = (int)(idx / cols), c = (int)(idx % cols);
  float di = dinv[i];
  float y = agg[(size_t)i * ld + c] + hlin[(size_t)i * ld + c] * di * di + bias[c];
  if (relu) y = fmaxf(y, 0.f);
  agg[(size_t)i * ld + c] = y;
}

// ---------------- batch norm ----------------
__global__ __launch_bounds__(256) void bn_stats_kernel(
    const float* __restrict__ X, int rows, int ld,
    float* __restrict__ mean, float* __restrict__ var) {
  int c = blockIdx.x;
  __shared__ float s1[256], s2[256];
  float a = 0.f, b = 0.f;
  for (int r = threadIdx.x; r < rows; r += 256) {
    float x = X[(size_t)r * ld + c];
    a += x; b += x * x;
  }
  s1[threadIdx.x] = a; s2[threadIdx.x] = b;
  __syncthreads();
  for (int s = 128; s > 0; s >>= 1) {
    if (threadIdx.x < s) { s1[threadIdx.x] += s1[threadIdx.x + s];
                           s2[threadIdx.x] += s2[threadIdx.x + s]; }
    __syncthreads();
  }
  if (threadIdx.x == 0) {
    float m = s1[0] / rows;
    mean[c] = m;
    var[c] = s2[0] / rows - m * m;
  }
}

__global__ void bn_apply_kernel(float* __restrict__ X,
                                const float* __restrict__ mean,
                                const float* __restrict__ var,
                                const float* __restrict__ g,
                                const float* __restrict__ be,
                                long long n, int cols, int ld, int relu) {
  long long idx = blockIdx.x * (long long)blockDim.x + threadIdx.x;
  if (idx >= n) return;
  int i = (int)(idx / cols), c = (int)(idx % cols);
  float y = (X[(size_t)i * ld + c] - mean[c]) * rsqrtf(var[c] + EPSBN) * g[c] + be[c];
  if (relu) y = fmaxf(y, 0.f);
  X[(size_t)i * ld + c] = y;
}

// ---------------- masked pooling: pool[b][0:35]=lig, [35:70]=pro ----------
__global__ void pool_kernel(const float* __restrict__ h,
                            const int* __restrict__ split,
                            const int* __restrict__ batch,
                            float* __restrict__ pool, int n, int ld) {
  int idx = blockIdx.x * blockDim.x + threadIdx.x;
  if (idx >= n * ND) return;
  int i = idx / ND, f = idx % ND;
  int isPro = (split[i] == 1) ? 1 : 0;
  atomicAdd(&pool[(size_t)batch[i] * (2 * ND) + isPro * ND + f],
            h[(size_t)i * ld + f]);
}

// ---------------- BiLSTM: one block per (direction, batch elem) ----------
__global__ __launch_bounds__(160) void lstm_kernel(
    const float* __restrict__ seq,   // [B,2,35]; t>=2 are zeros
    const float* __restrict__ Wih_f, const float* __restrict__ Whh_f,
    const float* __restrict__ bih_f, const float* __restrict__ bhh_f,
    const float* __restrict__ Wih_b, const float* __restrict__ Whh_b,
    const float* __restrict__ bih_b, const float* __restrict__ bhh_b,
    float* __restrict__ out)         // [B,140,70]
{
  const int b   = blockIdx.x % NB;
  const int dir = blockIdx.x / NB;
  const float* Wih = dir ? Wih_b : Wih_f;
  const float* Whh = dir ? Whh_b : Whh_f;
  const float* bih = dir ? bih_b : bih_f;
  const float* bhh = dir ? bhh_b : bhh_f;

  __shared__ float sWih[4 * ND * ND];
  __shared__ float sWhh[4 * ND * ND];
  __shared__ float h[ND], c[ND], gates[4 * ND], xt[ND];

  for (int i = threadIdx.x; i < 4 * ND * ND; i += blockDim.x) {
    sWih[i] = Wih[i]; sWhh[i] = Whh[i];
  }
  if (threadIdx.x < ND) { h[threadIdx.x] = 0.f; c[threadIdx.x] = 0.f; }
  __syncthreads();

  for (int t = 0; t < NT; ++t) {
    const int te = dir ? (NT - 1 - t) : t;
    if (threadIdx.x < ND)
      xt[threadIdx.x] = (te < 2) ? seq[((size_t)b * 2 + te) * ND + threadIdx.x] : 0.f;
    __syncthreads();
    const int g = threadIdx.x;
    if (g < 4 * ND) {
      float acc = bih[g] + bhh[g];
      const float* wi = &sWih[g * ND];
      const float* wh = &sWhh[g * ND];
      for (int k = 0; k < ND; ++k) acc += xt[k] * wi[k] + h[k] * wh[k];
      gates[g] = acc;
    }
    __syncthreads();
    if (threadIdx.x < ND) {
      const int j = threadIdx.x;
      float ig = 1.f / (1.f + __expf(-gates[j]));
      float fg = 1.f / (1.f + __expf(-gates[ND + j]));
      float gg = tanhf(gates[2 * ND + j]);
      float og = 1.f / (1.f + __expf(-gates[3 * ND + j]));
      float cn = fg * c[j] + ig * gg;
      float hn = og * tanhf(cn);
      c[j] = cn; h[j] = hn;
      out[((size_t)b * NT + te) * (2 * ND) + dir * ND + j] = hn;
    }
    __syncthreads();
  }
}

// ---------------- dense projection (q/k/v): Y = X @ W + b ----------------
__global__ void linproj_kernel(const float* __restrict__ X,
                               const float* __restrict__ W,
                               const float* __restrict__ bias,
                               float* __restrict__ Y,
                               int rows, int din, int dout) {
  int idx = blockIdx.x * blockDim.x + threadIdx.x;
  if (idx >= rows * dout) return;
  int r = idx / dout, c = idx % dout;
  const float* xp = X + (size_t)r * din;
  float acc = bias[c];
  for (int k = 0; k < din; ++k) acc += xp[k] * W[(size_t)k * dout + c];
  Y[idx] = acc;
}

// attention mask: replicate the .at[].set() sequence from the reference
__device__ __forceinline__ float attn_mask(int i, int j) {
  float v = (i == j) ? 1.f : 0.f;
  if (i >= 2) v = 0.f;
  if (j >= 2) v = 0.f;
  if (j == 1) v = 1.f;
  if (i == 1) v = 1.f;
  if (i == 1 && j == 1) v = 0.f;
  return v;
}

// ---------------- masked softmax attention: one block per (b, query) ------
// output row stride ldRow (fc1 input is [B, 9800] padded to OBLD)
__global__ __launch_bounds__(256) void attn_kernel(
    const float* __restrict__ q, const float* __restrict__ k,
    const float* __restrict__ v, float* __restrict__ o, int ldRow) {
  const int b  = blockIdx.x / NT;
  const int qi = blockIdx.x % NT;
  __shared__ float sc[NT];
  __shared__ float red[256];
  __shared__ float sq[2 * ND];
  const float* qp = q + ((size_t)b * NT + qi) * (2 * ND);
  if (threadIdx.x < 2 * ND) sq[threadIdx.x] = qp[threadIdx.x];
  __syncthreads();
  for (int j = threadIdx.x; j < NT; j += blockDim.x) {
    const float* kp = k + ((size_t)b * NT + j) * (2 * ND);
    float d = 0.f;
    for (int c = 0; c < 2 * ND; ++c) d += sq[c] * kp[c];
    d *= 0.11952286093343936f;          // 1/sqrt(70)
    if (attn_mask(qi, j) == 0.f) d = -1e9f;
    sc[j] = d;
  }
  __syncthreads();
  float m = -3.4e38f;
  for (int j = threadIdx.x; j < NT; j += blockDim.x) m = fmaxf(m, sc[j]);
  red[threadIdx.x] = m; __syncthreads();
  for (int s = 128; s > 0; s >>= 1) {
    if (threadIdx.x < s) red[threadIdx.x] = fmaxf(red[threadIdx.x], red[threadIdx.x + s]);
    __syncthreads();
  }
  m = red[0]; __syncthreads();
  float lsum = 0.f;
  for (int j = threadIdx.x; j < NT; j += blockDim.x) {
    float e = __expf(sc[j] - m); sc[j] = e; lsum += e;
  }
  red[threadIdx.x] = lsum; __syncthreads();
  for (int s = 128; s > 0; s >>= 1) {
    if (threadIdx.x < s) red[threadIdx.x] += red[threadIdx.x + s];
    __syncthreads();
  }
  const float inv = 1.f / red[0];
  __syncthreads();
  if (threadIdx.x < 2 * ND) {
    float acc = 0.f;
    for (int j = 0; j < NT; ++j)
      acc += sc[j] * v[((size_t)b * NT + j) * (2 * ND) + threadIdx.x];
    o[(size_t)b * ldRow + qi * (2 * ND) + threadIdx.x] = acc * inv;
  }
}

// ---------------- FC2 ----------------
__global__ __launch_bounds__(256) void fc2_kernel(
    const float* __restrict__ f, const float* __restrict__ W,
    const float* __restrict__ bias, float* __restrict__ out, int K) {
  const int b = blockIdx.x;
  __shared__ float red[256];
  float acc = 0.f;
  for (int k = threadIdx.x; k < K; k += 256) acc += f[(size_t)b * K + k] * W[k];
  red[threadIdx.x] = acc; __syncthreads();
  for (int s = 128; s > 0; s >>= 1) {
    if (threadIdx.x < s) red[threadIdx.x] += red[threadIdx.x + s];
    __syncthreads();
  }
  if (threadIdx.x == 0) out[b] = red[0] + bias[0];
}

// ================= host-side orchestration =================
static inline unsigned int nblk(long long total, int bs) {
  return (unsigned int)((total + bs - 1) / bs);
}

extern "C" void kernel_launch(void* const* d_in, const int* in_sizes, int n_in,
                              void* d_out, int out_size, void* d_ws, size_t ws_size,
                              hipStream_t stream) {
  (void)in_sizes; (void)n_in; (void)out_size; (void)ws_size;

  const float* x     = (const float*)d_in[0];
  const int*   ei    = (const int*)d_in[1];   // [2,300000]
  const int*   ee    = (const int*)d_in[2];   // [2,100000]
  const int*   split = (const int*)d_in[3];
  const int*   batch = (const int*)d_in[4];

  const float *lw[5], *lb[5], *lg[5], *lbe[5];
  for (int i = 0; i < 5; ++i) {
    lw[i]  = (const float*)d_in[5 + 4 * i + 0];
    lb[i]  = (const float*)d_in[5 + 4 * i + 1];
    lg[i]  = (const float*)d_in[5 + 4 * i + 2];
    lbe[i] = (const float*)d_in[5 + 4 * i + 3];
  }
  const float* Wih_f = (const float*)d_in[25];
  const float* Whh_f = (const float*)d_in[26];
  const float* bih_f = (const float*)d_in[27];
  const float* bhh_f = (const float*)d_in[28];
  const float* Wih_b = (const float*)d_in[29];
  const float* Whh_b = (const float*)d_in[30];
  const float* bih_b = (const float*)d_in[31];
  const float* bhh_b = (const float*)d_in[32];
  const float* Wq = (const float*)d_in[33];  const float* bq = (const float*)d_in[34];
  const float* Wk = (const float*)d_in[35];  const float* bk = (const float*)d_in[36];
  const float* Wv = (const float*)d_in[37];  const float* bv = (const float*)d_in[38];
  const float* Wfc1 = (const float*)d_in[39]; const float* bfc1 = (const float*)d_in[40];
  const float* Wfc2 = (const float*)d_in[41]; const float* bfc2 = (const float*)d_in[42];

  // ---- workspace carve-up (floats) ----
  float* ws = (float*)d_ws;
  float* B0   = ws;                              // MPAD*512
  float* B1   = B0 + (size_t)MPAD * 512;         // MPAD*512
  float* B2   = B1 + (size_t)MPAD * 512;         // MPAD*256
  float* xp   = B2 + (size_t)MPAD * 256;         // MPAD*48 (x, K pad 35->48, zeros)
  float* W5p  = xp + (size_t)MPAD * 48;          // 256*64  (N pad cols)
  float* deg  = W5p + 256 * 64;                  // MPAD (dinv)
  float* pool = deg + MPAD;                      // B*70 == seq [B,2,35]
  const size_t BT70 = (size_t)NB * NT * 2 * ND;
  float* lstm = pool + (size_t)NB * 2 * ND;      // [B,140,70]
  float* qb   = lstm + BT70;
  float* kb   = qb + BT70;
  float* vb   = kb + BT70;
  float* ob   = vb + BT70;                       // [B, OBLD], fc1 input (padded)
  float* fbuf = ob + (size_t)NB * OBLD;          // [B,4340]
  float* meanb = fbuf + (size_t)NB * 4340;       // 512
  float* varb  = meanb + 512;                    // 512

  const int BS = 256;

  // ---- pad x (K 35->48) and W5 (N 35->64) ----
  zero_kernel<<<nblk((long long)MPAD * 48, BS), BS, 0, stream>>>(xp, (long long)MPAD * 48);
  copy_pad_kernel<<<nblk((long long)NN * ND, BS), BS, 0, stream>>>(xp, x, NN, ND, 48);
  zero_kernel<<<nblk(256 * 64, BS), BS, 0, stream>>>(W5p, 256 * 64);
  copy_pad_kernel<<<nblk(256 * 35, BS), BS, 0, stream>>>(W5p, lw[4], 256, 35, 64);
  // ---- zero fc1-input pad columns ----
  zero_kernel<<<nblk((long long)NB * OBLD, BS), BS, 0, stream>>>(ob, (long long)NB * OBLD);

  // ---- degree / dinv (shared across layers) ----
  init_deg_kernel<<<nblk(NN, BS), BS, 0, stream>>>(deg, NN);
  count_deg_kernel<<<nblk(E_IN, BS), BS, 0, stream>>>(deg, ei + E_IN, E_IN);
  count_deg_kernel<<<nblk(E_EX, BS), BS, 0, stream>>>(deg, ee + E_EX, E_EX);
  finish_dinv_kernel<<<nblk(NN, BS), BS, 0, stream>>>(deg, NN);

  // ---- 5 GCN layers ----
  const int Kp  [5] = {48, 128, 256, 512, 256};  // padded K (A leading dim)
  const int Kr  [5] = {35, 128, 256, 512, 256};  // real K (B rows)
  const int Np  [5] = {128, 256, 512, 256, 64};  // padded N (= ld of outputs)
  const int cols[5] = {128, 256, 512, 256, ND};  // real feature counts
  const int preRelu [5] = {1, 1, 1, 0, 0};
  const int postRelu[5] = {0, 0, 0, 1, 0};
  const float* Wmat[5] = {lw[0], lw[1], lw[2], lw[3], W5p};
  const float* inp[5] = {xp, B1, B2, B0, B1};
  float*       lin[5] = {B0, B0, B1, B2, B2};
  float*       outp[5] = {B1, B2, B0, B1, B0};

  for (int L = 0; L < 5; ++L) {
    const int Nc = Np[L], cl = cols[L];
    // dense linear (guard-free WMMA path)
    dim3 ggrid(Nc / 64, MPAD / 64);
    gemm_wmma_tiled<false><<<ggrid, 128, 0, stream>>>(
        inp[L], Wmat[L], nullptr, lin[L], MPAD, Kp[L], Kr[L], Nc, 0);
    // aggregate
    zero_kernel<<<nblk((long long)MPAD * Nc, BS), BS, 0, stream>>>(
        outp[L], (long long)MPAD * Nc);
    scatter_kernel<<<nblk((long long)E_IN * cl, BS), BS, 0, stream>>>(
        lin[L], outp[L], ei, ei + E_IN, deg, (long long)E_IN * cl, cl, Nc);
    scatter_kernel<<<nblk((long long)E_EX * cl, BS), BS, 0, stream>>>(
        lin[L], outp[L], ee, ee + E_EX, deg, (long long)E_EX * cl, cl, Nc);
    combine_kernel<<<nblk((long long)NN * cl, BS), BS, 0, stream>>>(
        outp[L], lin[L], deg, lb[L], (long long)NN * cl, cl, Nc, preRelu[L]);
    // batch norm
    bn_stats_kernel<<<cl, 256, 0, stream>>>(outp[L], NN, Nc, meanb, varb);
    bn_apply_kernel<<<nblk((long long)NN * cl, BS), BS, 0, stream>>>(
        outp[L], meanb, varb, lg[L], lbe[L], (long long)NN * cl, cl, Nc, postRelu[L]);
  }
  float* hfin = outp[4];   // [N,35] with ld=64

  // ---- masked pooling -> seq [B,2,35] ----
  zero_kernel<<<nblk((long long)NB * 2 * ND, BS), BS, 0, stream>>>(
      pool, (long long)NB * 2 * ND);
  pool_kernel<<<nblk((long long)NN * ND, BS), BS, 0, stream>>>(
      hfin, split, batch, pool, NN, 64);

  // ---- BiLSTM ----
  lstm_kernel<<<2 * NB, 160, 0, stream>>>(pool, Wih_f, Whh_f, bih_f, bhh_f,
                                          Wih_b, Whh_b, bih_b, bhh_b, lstm);

  // ---- q/k/v projections ----
  const int rows = NB * NT;
  linproj_kernel<<<nblk((long long)rows * 2 * ND, BS), BS, 0, stream>>>(
      lstm, Wq, bq, qb, rows, 2 * ND, 2 * ND);
  linproj_kernel<<<nblk((long long)rows * 2 * ND, BS), BS, 0, stream>>>(
      lstm, Wk, bk, kb, rows, 2 * ND, 2 * ND);
  linproj_kernel<<<nblk((long long)rows * 2 * ND, BS), BS, 0, stream>>>(
      lstm, Wv, bv, vb, rows, 2 * ND, 2 * ND);

  // ---- masked attention (writes fc1 input with padded row stride) ----
  attn_kernel<<<NB * NT, 256, 0, stream>>>(qb, kb, vb, ob, OBLD);

  // ---- FC1 (WMMA, ragged N via index clamping): [128,9800]@[9800,4340] ----
  dim3 fgrid((4340 + 63) / 64, NB / 64);
  gemm_wmma_tiled<true><<<fgrid, 128, 0, stream>>>(
      ob, Wfc1, bfc1, fbuf, NB, OBLD, 9800, 4340, 1);

  // ---- FC2 ----
  fc2_kernel<<<NB, 256, 0, stream>>>(fbuf, Wfc2, bfc2, (float*)d_out, 4340);
}